// GRAFFNN_48309792145741
// MI455X (gfx1250) — compile-verified
//
#include <hip/hip_runtime.h>
#include <hip/hip_bf16.h>

typedef __attribute__((ext_vector_type(2))) float v2f;
typedef __attribute__((ext_vector_type(8))) float v8f;

#define NN      100000   // nodes
#define NE      1600000  // edges
#define KIN     128      // input feature dim
#define HD      64       // hidden dim == output dim
#define NLAYERS 4

// ---------------------------------------------------------------------------
// V_WMMA_F32_16X16X4_F32 : D(16x16 f32) = A(16x4 f32) * B(4x16 f32) + C
// A frag: lane<16 -> {A[m=lane][k0], A[m=lane][k0+1]}, lane>=16 -> k0+2,k0+3
// C/D:    VGPR r on lane L -> C[m = r + 8*(L>>4)][n = L&15]
// B frags are served from LDS in "paired-K" float2 layout so each fragment is
// ONE aligned ds_load_b64 into an even VGPR pair (no v_mov shuffling).
// ---------------------------------------------------------------------------
__device__ __forceinline__ v8f wmma4(v2f a, v2f b, v8f c) {
  return __builtin_amdgcn_wmma_f32_16x16x4_f32(
      /*neg_a=*/false, a, /*neg_b=*/false, b,
      /*c_mod=*/(short)0, c, /*reuse_a=*/false, /*reuse_b=*/false);
}

// ------------------------- graph normalization ----------------------------
__global__ void k_deg_init(float* deg) {
  int i = blockIdx.x * blockDim.x + threadIdx.x;
  if (i < NN) deg[i] = 1.0f;  // self-loop contribution
}

__global__ void k_deg_count(const int* __restrict__ ecol, float* __restrict__ deg) {
  int e = blockIdx.x * blockDim.x + threadIdx.x;
  if (e < NE) atomicAdd(&deg[ecol[e]], 1.0f);
}

__global__ void k_dinv(float* deg) {
  int i = blockIdx.x * blockDim.x + threadIdx.x;
  if (i < NN) deg[i] = rsqrtf(deg[i]);  // deg >= 1 always
}

// ------------------- symmetrize the three 64x64 mixers --------------------
// sym(W)[i][j] = (i<=j) ? W[i][j] : W[j][i]
__global__ void k_sym(const float* __restrict__ Wint, const float* __restrict__ Wext,
                      const float* __restrict__ Wini, float* __restrict__ Ws) {
  int idx = blockIdx.x * blockDim.x + threadIdx.x;  // 3 * 64 * 64
  if (idx >= 3 * HD * HD) return;
  int m = idx / (HD * HD);
  int ij = idx % (HD * HD);
  int i = ij / HD, j = ij % HD;
  const float* W = (m == 0) ? Wint : (m == 1) ? Wext : Wini;
  Ws[idx] = (i <= j) ? W[i * HD + j] : W[j * HD + i];
}

// ---- helper: stage row-major W[K][HD] into paired-K float2 LDS layout ----
// dst[(k>>1)*HD + n] = { W[k&~1][n], W[(k&~1)+1][n] }
__device__ __forceinline__ void stage_pairedK(float* dstF, const float* __restrict__ src,
                                              int K, int tid, int nthreads) {
  for (int i = tid; i < K * HD; i += nthreads) {
    int k = i / HD, n = i % HD;
    dstF[(((k >> 1) * HD + n) << 1) + (k & 1)] = src[i];
  }
}

// --------------------------- encoder GEMM ---------------------------------
// x0 = h = x @ W_enc + b_enc ; x: [NN,128], W_enc: [128,64]
__global__ __launch_bounds__(256)
void k_encoder(const float* __restrict__ x, const float* __restrict__ Wenc,
               const float* __restrict__ benc, float* __restrict__ x0,
               float* __restrict__ h) {
  __shared__ v2f sW[(KIN / 2) * HD];  // 32 KB, paired-K
  __shared__ float sb[HD];
  int tid = threadIdx.x;
  stage_pairedK((float*)sW, Wenc, KIN, tid, 256);
  if (tid < HD) sb[tid] = benc[tid];
  __syncthreads();

  int lane = tid & 31, wave = tid >> 5;
  int half = lane >> 4, mrow = lane & 15;
  int base = (blockIdx.x * 8 + wave) * 16;
  if (base >= NN) return;  // whole-wave exit: EXEC stays all-1s for WMMA

  const float* A = x + (size_t)base * KIN;
  v8f acc[4] = {};
  for (int k0 = 0; k0 < KIN; k0 += 4) {
    v2f a = *(const v2f*)(A + mrow * KIN + k0 + 2 * half);   // global_load_b64
    int prow = ((k0 >> 1) + half) * HD;
#pragma unroll
    for (int nt = 0; nt < 4; ++nt) {
      v2f b = sW[prow + nt * 16 + mrow];                     // ds_load_b64
      acc[nt] = wmma4(a, b, acc[nt]);
    }
  }
#pragma unroll
  for (int nt = 0; nt < 4; ++nt) {
    int nc = nt * 16 + mrow;
    float bias = sb[nc];
#pragma unroll
    for (int r = 0; r < 8; ++r) {
      float v = acc[nt][r] + bias;
      size_t off = (size_t)(base + half * 8 + r) * HD + nc;
      x0[off] = v;
      h[off]  = v;
    }
  }
}

// ------------------ fused per-layer pre-computation -----------------------
// t   = h @ Wi
// agg = dinv[i]^2 * t[i]  -  h @ We  -  x0 @ W0     (self-loop msg folded in)
__global__ __launch_bounds__(256)
void k_layer_pre(const float* __restrict__ h, const float* __restrict__ x0,
                 const float* __restrict__ Ws, const float* __restrict__ dinv,
                 float* __restrict__ t, float* __restrict__ agg) {
  __shared__ v2f sWi[(HD / 2) * HD];
  __shared__ v2f sWe[(HD / 2) * HD];
  __shared__ v2f sW0[(HD / 2) * HD];  // 48 KB total, paired-K
  int tid = threadIdx.x;
  stage_pairedK((float*)sWi, Ws,               HD, tid, 256);
  stage_pairedK((float*)sWe, Ws + HD * HD,     HD, tid, 256);
  stage_pairedK((float*)sW0, Ws + 2 * HD * HD, HD, tid, 256);
  __syncthreads();

  int lane = tid & 31, wave = tid >> 5;
  int half = lane >> 4, mrow = lane & 15;
  int base = (blockIdx.x * 8 + wave) * 16;
  if (base >= NN) return;

  const float* Ah = h  + (size_t)base * HD;
  const float* Ax = x0 + (size_t)base * HD;
  v8f at[4] = {};  // h @ Wi
  v8f au[4] = {};  // h @ We + x0 @ W0
  for (int k0 = 0; k0 < HD; k0 += 4) {
    v2f ah = *(const v2f*)(Ah + mrow * HD + k0 + 2 * half);
    v2f ax = *(const v2f*)(Ax + mrow * HD + k0 + 2 * half);
    int prow = ((k0 >> 1) + half) * HD;
#pragma unroll
    for (int nt = 0; nt < 4; ++nt) {
      int pc = prow + nt * 16 + mrow;
      at[nt] = wmma4(ah, sWi[pc], at[nt]);
      au[nt] = wmma4(ah, sWe[pc], au[nt]);
      au[nt] = wmma4(ax, sW0[pc], au[nt]);
    }
  }

  float d2[8];
#pragma unroll
  for (int r = 0; r < 8; ++r) {
    float d = dinv[base + half * 8 + r];
    d2[r] = d * d;  // self-loop norm
  }
#pragma unroll
  for (int nt = 0; nt < 4; ++nt) {
    int nc = nt * 16 + mrow;
#pragma unroll
    for (int r = 0; r < 8; ++r) {
      size_t off = (size_t)(base + half * 8 + r) * HD + nc;
      float tv = at[nt][r];
      t[off]   = tv;
      agg[off] = tv * d2[r] - au[nt][r];
    }
  }
}

// ------------------------- edge gather/scatter ----------------------------
// agg[col] += t[row] * dinv[row] * dinv[col] ; 16 threads per edge, float4 each
__global__ __launch_bounds__(256)
void k_edge(const int* __restrict__ erow, const int* __restrict__ ecol,
            const float* __restrict__ dinv, const float* __restrict__ t,
            float* __restrict__ agg) {
  size_t idx = (size_t)blockIdx.x * blockDim.x + threadIdx.x;
  size_t e = idx >> 4;
  if (e >= NE) return;
  int c  = (int)(idx & 15) << 2;
  int r  = erow[e];
  int cl = ecol[e];
  float w = dinv[r] * dinv[cl];
  const float4 v = *(const float4*)(t + (size_t)r * HD + c);
  float* ag = agg + (size_t)cl * HD + c;
  atomicAdd(ag + 0, v.x * w);
  atomicAdd(ag + 1, v.y * w);
  atomicAdd(ag + 2, v.z * w);
  atomicAdd(ag + 3, v.w * w);
}

// --------------------------- layer update ---------------------------------
// h = relu(h + 1.0 * relu(agg))
__global__ __launch_bounds__(256)
void k_update(float* __restrict__ h, const float* __restrict__ agg) {
  size_t i = (size_t)blockIdx.x * blockDim.x + threadIdx.x;
  if (i >= (size_t)NN * HD / 4) return;
  float4 hv = ((float4*)h)[i];
  float4 av = ((const float4*)agg)[i];
  hv.x = fmaxf(hv.x + fmaxf(av.x, 0.f), 0.f);
  hv.y = fmaxf(hv.y + fmaxf(av.y, 0.f), 0.f);
  hv.z = fmaxf(hv.z + fmaxf(av.z, 0.f), 0.f);
  hv.w = fmaxf(hv.w + fmaxf(av.w, 0.f), 0.f);
  ((float4*)h)[i] = hv;
}

// --------------------------- decoder GEMM ---------------------------------
__global__ __launch_bounds__(256)
void k_decoder(const float* __restrict__ h, const float* __restrict__ Wdec,
               const float* __restrict__ bdec, float* __restrict__ out) {
  __shared__ v2f sW[(HD / 2) * HD];
  __shared__ float sb[HD];
  int tid = threadIdx.x;
  stage_pairedK((float*)sW, Wdec, HD, tid, 256);
  if (tid < HD) sb[tid] = bdec[tid];
  __syncthreads();

  int lane = tid & 31, wave = tid >> 5;
  int half = lane >> 4, mrow = lane & 15;
  int base = (blockIdx.x * 8 + wave) * 16;
  if (base >= NN) return;

  const float* A = h + (size_t)base * HD;
  v8f acc[4] = {};
  for (int k0 = 0; k0 < HD; k0 += 4) {
    v2f a = *(const v2f*)(A + mrow * HD + k0 + 2 * half);
    int prow = ((k0 >> 1) + half) * HD;
#pragma unroll
    for (int nt = 0; nt < 4; ++nt) {
      v2f b = sW[prow + nt * 16 + mrow];
      acc[nt] = wmma4(a, b, acc[nt]);
    }
  }
#pragma unroll
  for (int nt = 0; nt < 4; ++nt) {
    int nc = nt * 16 + mrow;
    float bias = sb[nc];
#pragma unroll
    for (int r = 0; r < 8; ++r)
      out[(size_t)(base + half * 8 + r) * HD + nc] = acc[nt][r] + bias;
  }
}

// ---------------------------------------------------------------------------
extern "C" void kernel_launch(void* const* d_in, const int* in_sizes, int n_in,
                              void* d_out, int out_size, void* d_ws, size_t ws_size,
                              hipStream_t stream) {
  const float* x     = (const float*)d_in[0];
  const int*   eidx  = (const int*)d_in[1];   // (2, E): row then col
  const float* Wenc  = (const float*)d_in[2];
  const float* benc  = (const float*)d_in[3];
  const float* Wint  = (const float*)d_in[4];
  const float* Wext  = (const float*)d_in[5];
  const float* Wini  = (const float*)d_in[6];
  const float* Wdec  = (const float*)d_in[7];
  const float* bdec  = (const float*)d_in[8];
  float* out = (float*)d_out;

  const int* erow = eidx;
  const int* ecol = eidx + NE;

  // workspace layout (floats): dinv | x0 | h | t | agg | Ws(3x64x64)
  float* ws   = (float*)d_ws;
  float* dinv = ws;                       // NN
  float* x0   = dinv + NN;                // NN*HD
  float* h    = x0 + (size_t)NN * HD;     // NN*HD
  float* t    = h  + (size_t)NN * HD;     // NN*HD
  float* agg  = t  + (size_t)NN * HD;     // NN*HD
  float* Ws   = agg + (size_t)NN * HD;    // 3*HD*HD

  const int TB = 256;
  const int gN   = (NN + TB - 1) / TB;
  const int gE   = (NE + TB - 1) / TB;
  const int gGEMM = (NN + 127) / 128;                 // 8 waves x 16 rows / block
  const int gEdge = (int)(((size_t)NE * 16 + TB - 1) / TB);
  const int gUpd  = (int)(((size_t)NN * HD / 4 + TB - 1) / TB);

  // 1) gcn_norm: deg (with self loops) -> dinv
  k_deg_init<<<gN, TB, 0, stream>>>(dinv);
  k_deg_count<<<gE, TB, 0, stream>>>(ecol, dinv);
  k_dinv<<<gN, TB, 0, stream>>>(dinv);

  // 2) symmetric mixers
  k_sym<<<(3 * HD * HD + TB - 1) / TB, TB, 0, stream>>>(Wint, Wext, Wini, Ws);

  // 3) encoder
  k_encoder<<<gGEMM, TB, 0, stream>>>(x, Wenc, benc, x0, h);

  // 4) GRAFF layers
  for (int l = 0; l < NLAYERS; ++l) {
    k_layer_pre<<<gGEMM, TB, 0, stream>>>(h, x0, Ws, dinv, t, agg);
    k_edge<<<gEdge, TB, 0, stream>>>(erow, ecol, dinv, t, agg);
    k_update<<<gUpd, TB, 0, stream>>>(h, agg);
  }

  // 5) decoder
  k_decoder<<<gGEMM, TB, 0, stream>>>(h, Wdec, bdec, out);
}